// GeoNew_86577950753083
// MI455X (gfx1250) — compile-verified
//
#include <hip/hip_runtime.h>
#include <hip/hip_bf16.h>
#include <math.h>

#define BB 16
#define NN 1024
#define DIN 32
#define DZ 64
#define DH 64
#define FF 3
#define PP_P 19
#define PPAD 32
#define NBC 3
#define NDIR 64
#define NEDGE 171
#define MAXIT 50
#define ALPHA 0.1f
#define AS 1028  // LDS A-panel row stride in words; 1028 % 64 == 4 -> conflict-free b64 reads

typedef float v2f __attribute__((ext_vector_type(2)));
typedef float v8f __attribute__((ext_vector_type(8)));

// ---------------- init: zero bvals_full and K_hat/M_hat accumulators ----------------
__global__ void k_init_zero(float* bvals, float* khat) {
    int i = blockIdx.x * blockDim.x + threadIdx.x;
    if (i < BB * NN * FF) bvals[i] = 0.f;
    if (i < BB * FF * 2 * PPAD * PPAD) khat[i] = 0.f;
}

// ---------------- scatter boundary_vals at dirichlet nodes ----------------
__global__ void k_scatter(const int* dn, const float* bv, float* bvals) {
    int t = blockIdx.x * blockDim.x + threadIdx.x;
    if (t >= BB * NDIR * FF) return;
    int f = t % FF, k = (t / FF) % NDIR, b = t / (FF * NDIR);
    int node = dn[b * NDIR + k];
    bvals[(b * NN + node) * FF + f] = bv[(b * NDIR + k) * FF + f];
}

// ---------------- encoder: z = tanh(tok @ W_enc + b_enc) ----------------
__global__ void k_encode(const float* __restrict__ tok, const float* __restrict__ We,
                         const float* __restrict__ be, float* __restrict__ z) {
    int t = blockIdx.x * blockDim.x + threadIdx.x;
    if (t >= BB * NN * DZ) return;
    int d = t % DZ;
    int bi = t / DZ;  // b*N+i
    const float* x = tok + (size_t)bi * DIN;
    float acc = be[d];
#pragma unroll
    for (int c = 0; c < DIN; ++c) acc += x[c] * We[c * DZ + d];
    z[t] = tanhf(acc);
}

// ---------------- POU softmax weights -> Wbuf [B,F,PPAD,N] ----------------
__global__ void k_pou(const float* __restrict__ z, const float* __restrict__ Wp,
                      const float* __restrict__ bvals, float* __restrict__ Wbuf) {
    int t = blockIdx.x * blockDim.x + threadIdx.x;
    if (t >= BB * NN * FF) return;
    int f = t % FF, i = (t / FF) % NN, b = t / (FF * NN);
    const float* zv = z + ((size_t)b * NN + i) * DZ;
    float logit[PP_P];
    for (int p = 0; p < PP_P; ++p) {
        float acc = 0.f;
        for (int d = 0; d < DZ; ++d) acc += zv[d] * Wp[d * (PP_P * FF) + p * FF + f];
        logit[p] = acc;
    }
    float bvf = bvals[((size_t)b * NN + i) * FF + f];
    for (int p = PP_P - NBC; p < PP_P; ++p) logit[p] += bvf;
    float mx = logit[0];
    for (int p = 1; p < PP_P; ++p) mx = fmaxf(mx, logit[p]);
    float s = 0.f;
    for (int p = 0; p < PP_P; ++p) { logit[p] = __expf(logit[p] - mx); s += logit[p]; }
    float inv = 1.f / s;
    float* wout = Wbuf + ((size_t)(b * FF + f) * PPAD) * NN + i;
    for (int p = 0; p < PP_P; ++p) wout[(size_t)p * NN] = logit[p] * inv;
    for (int p = PP_P; p < PPAD; ++p) wout[(size_t)p * NN] = 0.f;  // zero pad rows
}

// ---------------- f_full = z @ W_src + b_src ----------------
__global__ void k_ffull(const float* __restrict__ z, const float* __restrict__ Ws,
                        const float* __restrict__ bs, float* __restrict__ ff) {
    int t = blockIdx.x * blockDim.x + threadIdx.x;
    if (t >= BB * NN * FF) return;
    int f = t % FF;
    int bi = t / FF;
    const float* zv = z + (size_t)bi * DZ;
    float acc = bs[f];
    for (int d = 0; d < DZ; ++d) acc += zv[d] * Ws[d * FF + f];
    ff[t] = acc;
}

// ---------------- zmean then zp = mean(z) @ Wz ----------------
__global__ void k_zmean(const float* __restrict__ z, float* __restrict__ zm) {
    int t = blockIdx.x * blockDim.x + threadIdx.x;
    if (t >= BB * DZ) return;
    int d = t % DZ, b = t / DZ;
    float acc = 0.f;
    const float* zb = z + (size_t)b * NN * DZ + d;
    for (int i = 0; i < NN; ++i) acc += zb[(size_t)i * DZ];
    zm[t] = acc * (1.0f / NN);
}
__global__ void k_zp(const float* __restrict__ zm, const float* __restrict__ Wz,
                     float* __restrict__ zp) {
    int t = blockIdx.x * blockDim.x + threadIdx.x;
    if (t >= BB * DH) return;
    int h = t % DH, b = t / DH;
    float acc = 0.f;
    for (int d = 0; d < DZ; ++d) acc += zm[b * DZ + d] * Wz[d * DH + h];
    zp[t] = acc;
}

// ---------------- projection: K_hat = (W K W^T)/scal via fp32 WMMA ----------------
// One block = (b,f,mat,jgroup-of-8). A-panel (32 x 1024 POU rows) staged once in LDS,
// shared by 8 waves (one 16-col j-tile each). Each wave computes BOTH 16-row M-tiles
// with two accumulators so the streaming K/M operand is read exactly once (NT hint).
__global__ void __launch_bounds__(256) k_project(const float* __restrict__ Kl,
                                                 const float* __restrict__ Ml,
                                                 const float* __restrict__ Wbuf,
                                                 float* __restrict__ khat) {
    __shared__ float Albuf[PPAD * AS];       // ~128.5 KB padded A panel
    __shared__ float tbuf[8][16][17];        // per-wave T-tile spill
    const int tid = threadIdx.x;
    const int wave = tid >> 5;
    const int lane = tid & 31;
    int id = blockIdx.x;                     // 768 blocks total
    const int jg = id & 7;  id >>= 3;
    const int mat = id & 1; id >>= 1;
    const int f = id % 3;
    const int b = id / 3;

    const float* Ksrc = (mat ? Ml : Kl) + (size_t)(b * FF + f) * NN * NN;
    const float* Wpan = Wbuf + (size_t)(b * FF + f) * PPAD * NN;

    // cooperative A-panel preload (float4 chunks, 16B-aligned in LDS: AS*4 = 257*16)
    for (int idx = tid; idx < PPAD * (NN / 4); idx += 256) {
        int row = idx >> 8;                  // 256 float4 per row
        int c4 = (idx & 255) << 2;
        float4 v = *(const float4*)(Wpan + (size_t)row * NN + c4);
        *(float4*)&Albuf[row * AS + c4] = v;
    }
    __syncthreads();

    const int m = lane & 15;                 // A row (local) == B col (local)
    const int hi = lane >> 4;
    const int ksel = hi * 2;                 // K-striping within a 4-chunk
    const int jt = jg * 8 + wave;
    const int j0 = jt * 16;

    v8f acc0 = {}, acc1 = {};
    const float* a0p = &Albuf[m * AS];
    const float* a1p = &Albuf[(m + 16) * AS];
    for (int kk = 0; kk < NN; kk += 4) {
        v2f a0, a1, bv;
        a0.x = a0p[kk + ksel];
        a0.y = a0p[kk + ksel + 1];
        a1.x = a1p[kk + ksel];
        a1.y = a1p[kk + ksel + 1];
        bv.x = __builtin_nontemporal_load(&Ksrc[(size_t)(kk + ksel) * NN + j0 + m]);
        bv.y = __builtin_nontemporal_load(&Ksrc[(size_t)(kk + ksel + 1) * NN + j0 + m]);
        acc0 = __builtin_amdgcn_wmma_f32_16x16x4_f32(false, a0, false, bv,
                                                     (short)0, acc0, false, false);
        acc1 = __builtin_amdgcn_wmma_f32_16x16x4_f32(false, a1, false, bv,
                                                     (short)0, acc1, false, false);
    }

    // stage 2: K_hat[p,q] += sum_n T[p,n] * W[q, j0+n]  (B2 read from resident A panel)
    const float inv_scal = (float)PP_P / (float)NN;
    float* outbase = khat + (size_t)((b * FF + f) * 2 + mat) * PPAD * PPAD;
#pragma unroll
    for (int mt = 0; mt < 2; ++mt) {
        v8f accT = mt ? acc1 : acc0;
        __syncthreads();                     // uniform; protects tbuf reuse across mt
#pragma unroll
        for (int rr = 0; rr < 8; ++rr) tbuf[wave][rr + 8 * hi][m] = accT[rr];
        __syncthreads();
#pragma unroll
        for (int qt = 0; qt < 2; ++qt) {
            v8f acc2 = {};
#pragma unroll
            for (int c = 0; c < 4; ++c) {
                v2f a2, b2;
                a2.x = tbuf[wave][m][4 * c + ksel];
                a2.y = tbuf[wave][m][4 * c + ksel + 1];
                int qrow = qt * 16 + m;      // global q row of W panel
                b2.x = Albuf[qrow * AS + j0 + 4 * c + ksel];
                b2.y = Albuf[qrow * AS + j0 + 4 * c + ksel + 1];
                acc2 = __builtin_amdgcn_wmma_f32_16x16x4_f32(false, a2, false, b2,
                                                             (short)0, acc2, false, false);
            }
#pragma unroll
            for (int rr = 0; rr < 8; ++rr) {
                int p = mt * 16 + rr + 8 * hi;
                atomicAdd(&outbase[(size_t)p * PPAD + qt * 16 + m], acc2[rr] * inv_scal);
            }
        }
    }
}

// ---------------- Dirichlet rows -> identity ----------------
__global__ void k_dirichlet(float* khat) {
    int t = blockIdx.x * blockDim.x + threadIdx.x;
    if (t >= BB * FF * 2 * NBC * PPAD) return;
    int q = t % PPAD, pr = (t / PPAD) % NBC, mtx = t / (PPAD * NBC);
    int p = PP_P - NBC + pr;
    khat[(size_t)mtx * PPAD * PPAD + p * PPAD + q] = (q == p) ? 1.f : 0.f;
}

// ---------------- f_latent = W @ f_full / scal ----------------
__global__ void k_flatent(const float* __restrict__ Wbuf, const float* __restrict__ ff,
                          float* __restrict__ fl) {
    int t = blockIdx.x * blockDim.x + threadIdx.x;
    if (t >= BB * PP_P * FF) return;
    int f = t % FF, p = (t / FF) % PP_P, b = t / (FF * PP_P);
    const float* wr = Wbuf + ((size_t)(b * FF + f) * PPAD + p) * NN;
    float acc = 0.f;
    for (int i = 0; i < NN; ++i) acc += wr[i] * ff[((size_t)b * NN + i) * FF + f];
    fl[(b * PPAD + p) * FF + f] = acc * ((float)PP_P / (float)NN);
}

// ---------------- src_term = M_hat @ f_latent ----------------
__global__ void k_src(const float* __restrict__ khat, const float* __restrict__ fl,
                      float* __restrict__ st) {
    int t = blockIdx.x * blockDim.x + threadIdx.x;
    if (t >= BB * PP_P * FF) return;
    int f = t % FF, p = (t / FF) % PP_P, b = t / (FF * PP_P);
    const float* Mh = khat + (size_t)((b * FF + f) * 2 + 1) * PPAD * PPAD;
    float acc = 0.f;
    for (int q = 0; q < PP_P; ++q) acc += Mh[p * PPAD + q] * fl[(b * PPAD + q) * FF + f];
    st[(b * PPAD + p) * FF + f] = acc;
}

// ---------------- 50-iteration damped fixed-point solver, one block per batch ----------
__global__ void __launch_bounds__(192) k_solve(const float* __restrict__ khat,
                                               const float* __restrict__ st,
                                               const float* __restrict__ zp,
                                               const float* __restrict__ uin,
                                               const float* __restrict__ Wf1,
                                               const float* __restrict__ bfv,
                                               const float* __restrict__ Wf2,
                                               float* __restrict__ uout) {
    const int b = blockIdx.x;
    const int tid = threadIdx.x;
    const int nthr = blockDim.x;
    __shared__ float Kl[FF * PPAD * PPAD];
    __shared__ float u_s[PPAD * FF], flux_s[PPAD * FF], r_s[PPAD * FF], st_s[PPAD * FF];
    __shared__ float zp_s[DH], bf_s[DH], W1_s[2 * FF * DH], W2_s[DH * FF];
    __shared__ unsigned char ei_s[NEDGE], ej_s[NEDGE];

    for (int i = tid; i < FF * PPAD * PPAD; i += nthr) {
        int f = i / (PPAD * PPAD), pq = i % (PPAD * PPAD);
        Kl[i] = khat[(size_t)((b * FF + f) * 2 + 0) * PPAD * PPAD + pq];
    }
    for (int i = tid; i < DH; i += nthr) { zp_s[i] = zp[b * DH + i]; bf_s[i] = bfv[i]; }
    for (int i = tid; i < 2 * FF * DH; i += nthr) W1_s[i] = Wf1[i];
    for (int i = tid; i < DH * FF; i += nthr) W2_s[i] = Wf2[i];
    for (int i = tid; i < PPAD * FF; i += nthr) { st_s[i] = st[b * PPAD * FF + i]; u_s[i] = 0.f; }
    for (int i = tid; i < PP_P * FF; i += nthr) u_s[i] = uin[b * (PP_P * FF) + i];
    if (tid == 0) {
        int e = 0;
        for (int i = 0; i < PP_P; ++i)
            for (int j = i + 1; j < PP_P; ++j) { ei_s[e] = (unsigned char)i; ej_s[e] = (unsigned char)j; ++e; }
    }
    __syncthreads();

    for (int it = 0; it < MAXIT; ++it) {
        for (int i = tid; i < PPAD * FF; i += nthr) flux_s[i] = 0.f;
        __syncthreads();
        for (int e = tid; e < NEDGE; e += nthr) {
            int i = ei_s[e], j = ej_s[e];
            float ue[6] = { u_s[i * FF + 0], u_s[i * FF + 1], u_s[i * FF + 2],
                            u_s[j * FF + 0], u_s[j * FF + 1], u_s[j * FF + 2] };
            float fl0 = 0.f, fl1 = 0.f, fl2 = 0.f;
            for (int h = 0; h < DH; ++h) {
                float a = bf_s[h] + zp_s[h];
#pragma unroll
                for (int c = 0; c < 6; ++c) a += ue[c] * W1_s[c * DH + h];
                float th = tanhf(a);
                fl0 += th * W2_s[h * FF + 0];
                fl1 += th * W2_s[h * FF + 1];
                fl2 += th * W2_s[h * FF + 2];
            }
            atomicAdd(&flux_s[i * FF + 0], -fl0); atomicAdd(&flux_s[j * FF + 0], fl0);
            atomicAdd(&flux_s[i * FF + 1], -fl1); atomicAdd(&flux_s[j * FF + 1], fl1);
            atomicAdd(&flux_s[i * FF + 2], -fl2); atomicAdd(&flux_s[j * FF + 2], fl2);
        }
        __syncthreads();
        if (tid < PP_P * FF) {
            int p = tid / FF, f = tid % FF;
            float r;
            if (p >= PP_P - NBC) {
                r = u_s[tid] - ((p == PP_P - NBC) ? 1.f : 0.f);
            } else {
                float acc = 0.f;
                for (int q = 0; q < PP_P; ++q)
                    acc += Kl[f * PPAD * PPAD + p * PPAD + q] * u_s[q * FF + f];
                r = acc - st_s[tid] - flux_s[tid];
            }
            r_s[tid] = r;
        }
        __syncthreads();
        if (tid < PP_P * FF) u_s[tid] -= ALPHA * r_s[tid];
        __syncthreads();
    }
    for (int i = tid; i < PP_P * FF; i += nthr) uout[b * PPAD * FF + i] = u_s[i];
}

// ---------------- u_fine = W^T @ u_coarse -> d_out [B,N,F] ----------------
__global__ void k_ufine(const float* __restrict__ Wbuf, const float* __restrict__ u,
                        float* __restrict__ out) {
    int t = blockIdx.x * blockDim.x + threadIdx.x;
    if (t >= BB * NN * FF) return;
    int f = t % FF, i = (t / FF) % NN, b = t / (FF * NN);
    const float* wr = Wbuf + ((size_t)(b * FF + f) * PPAD) * NN + i;
    const float* ub = u + b * PPAD * FF + f;
    float acc = 0.f;
    for (int p = 0; p < PP_P; ++p) acc += wr[(size_t)p * NN] * ub[p * FF];
    out[t] = acc;
}

extern "C" void kernel_launch(void* const* d_in, const int* in_sizes, int n_in,
                              void* d_out, int out_size, void* d_ws, size_t ws_size,
                              hipStream_t stream) {
    const float* in_tokens = (const float*)d_in[0];
    const float* K_list    = (const float*)d_in[1];
    const float* M_list    = (const float*)d_in[2];
    const int*   dnodes    = (const int*)d_in[3];
    const float* bvals_in  = (const float*)d_in[4];
    const float* u_init    = (const float*)d_in[5];
    const float* W_enc     = (const float*)d_in[6];
    const float* b_enc     = (const float*)d_in[7];
    const float* W_src     = (const float*)d_in[8];
    const float* b_src     = (const float*)d_in[9];
    const float* W_pou     = (const float*)d_in[10];
    const float* Wf1       = (const float*)d_in[11];
    const float* Wz        = (const float*)d_in[12];
    const float* bf        = (const float*)d_in[13];
    const float* Wf2       = (const float*)d_in[14];
    float* out = (float*)d_out;

    float* ws = (float*)d_ws;
    size_t off = 0;
    float* z     = ws + off; off += (size_t)BB * NN * DZ;              // 1,048,576
    float* Wbuf  = ws + off; off += (size_t)BB * FF * PPAD * NN;       // 1,572,864
    float* ffull = ws + off; off += (size_t)BB * NN * FF;              // 49,152
    float* bvals = ws + off; off += (size_t)BB * NN * FF;              // 49,152
    float* khat  = ws + off; off += (size_t)BB * FF * 2 * PPAD * PPAD; // 98,304
    float* zm    = ws + off; off += (size_t)BB * DZ;
    float* zp    = ws + off; off += (size_t)BB * DH;
    float* flat  = ws + off; off += (size_t)BB * PPAD * FF;
    float* stm   = ws + off; off += (size_t)BB * PPAD * FF;
    float* uhat  = ws + off; off += (size_t)BB * PPAD * FF;

    const int T = 256;
    k_init_zero<<<(BB * FF * 2 * PPAD * PPAD + T - 1) / T, T, 0, stream>>>(bvals, khat);
    k_scatter<<<(BB * NDIR * FF + T - 1) / T, T, 0, stream>>>(dnodes, bvals_in, bvals);
    k_encode<<<(BB * NN * DZ + T - 1) / T, T, 0, stream>>>(in_tokens, W_enc, b_enc, z);
    k_pou<<<(BB * NN * FF + T - 1) / T, T, 0, stream>>>(z, W_pou, bvals, Wbuf);
    k_ffull<<<(BB * NN * FF + T - 1) / T, T, 0, stream>>>(z, W_src, b_src, ffull);
    k_zmean<<<(BB * DZ + T - 1) / T, T, 0, stream>>>(z, zm);
    k_zp<<<(BB * DH + T - 1) / T, T, 0, stream>>>(zm, Wz, zp);
    // 16*3*2*8 = 768 blocks, 8 waves each; A-panel LDS-resident, K/M streamed once (NT)
    k_project<<<BB * FF * 2 * 8, 256, 0, stream>>>(K_list, M_list, Wbuf, khat);
    k_dirichlet<<<(BB * FF * 2 * NBC * PPAD + T - 1) / T, T, 0, stream>>>(khat);
    k_flatent<<<(BB * PP_P * FF + T - 1) / T, T, 0, stream>>>(Wbuf, ffull, flat);
    k_src<<<(BB * PP_P * FF + T - 1) / T, T, 0, stream>>>(khat, flat, stm);
    k_solve<<<BB, 192, 0, stream>>>(khat, stm, zp, u_init, Wf1, bf, Wf2, uhat);
    k_ufine<<<(BB * NN * FF + T - 1) / T, T, 0, stream>>>(Wbuf, uhat, out);
}